// JetGAT_85383949844946
// MI455X (gfx1250) — compile-verified
//
#include <hip/hip_runtime.h>
#include <math.h>

#define NN 100000            // nodes
#define NE 1600000           // edges (without self loops)
#define ET (NE + NN)         // edges + self loops
#define NG 1024              // graphs
#define EPS_BN 1e-5f

typedef float v2f __attribute__((ext_vector_type(2)));
typedef float v8f __attribute__((ext_vector_type(8)));

__device__ __forceinline__ float eluf(float x)   { return x > 0.f ? x : (expf(x) - 1.f); }
__device__ __forceinline__ float lrelu02(float x){ return x > 0.f ? x : 0.2f * x; }

// float atomic max via sign-split integer atomics (works for mixed signs, -inf init)
__device__ __forceinline__ void atomicMaxF(float* addr, float val) {
  if (val >= 0.f) atomicMax((int*)addr, __float_as_int(val));
  else            atomicMin((unsigned int*)addr, __float_as_uint(val));
}

__global__ void fill_kernel(float* __restrict__ p, float v, int n) {
  int i = blockIdx.x * blockDim.x + threadIdx.x;
  if (i < n) p[i] = v;
}

// ---------------- conv1 linear: h = x @ W1   (NN x 5)(5 x 128) ----------------
__global__ __launch_bounds__(128) void lin1_kernel(const float* __restrict__ x,
                                                   const float* __restrict__ W,
                                                   float* __restrict__ h) {
  int n = blockIdx.x, c = threadIdx.x;
  const float* xr = x + n * 5;
  float acc = 0.f;
  #pragma unroll
  for (int k = 0; k < 5; ++k) acc += xr[k] * W[k * 128 + c];
  h[(size_t)n * 128 + c] = acc;
}

// ---------------- fp32 WMMA GEMM: C[M,Nout] = A[M,K] @ B[K,Nout] ----------------
// One wave computes one 16x16 tile via V_WMMA_F32_16X16X4_F32. M % 16 == 0.
__global__ __launch_bounds__(32) void wmma_gemm_kernel(const float* __restrict__ A,
                                                       const float* __restrict__ B,
                                                       float* __restrict__ C,
                                                       int K, int Nout) {
  int tm = blockIdx.x, tn = blockIdx.y;
  int lane = threadIdx.x;
  int half = lane >> 4;          // lanes 16..31 hold K+2,K+3 (ISA 7.12.2)
  int l    = lane & 15;
  const float* Ar = A + (size_t)(tm * 16 + l) * K;
  const float* Bc = B + tn * 16 + l;
  v8f acc = {};
  for (int k = 0; k < K; k += 4) {
    v2f a, b;
    a[0] = Ar[k + 2 * half + 0];
    a[1] = Ar[k + 2 * half + 1];
    b[0] = Bc[(size_t)(k + 2 * half + 0) * Nout];
    b[1] = Bc[(size_t)(k + 2 * half + 1) * Nout];
    acc = __builtin_amdgcn_wmma_f32_16x16x4_f32(false, a, false, b, (short)0, acc,
                                                false, false);
  }
  // C/D layout: VGPR r -> row (r + 8*half), col = lane&15
  float* Cr = C + (size_t)(tm * 16 + 8 * half) * Nout + tn * 16 + l;
  #pragma unroll
  for (int r = 0; r < 8; ++r) Cr[(size_t)r * Nout] = acc[r];
}

// ---------------- per-node attention coefficients ----------------
// al_s[n,h] = sum_c h[n,h,c]*a_src[h,c]  (and same for a_dst)
template <int H, int C>
__global__ void attn_kernel(const float* __restrict__ h, const float* __restrict__ a_src,
                            const float* __restrict__ a_dst, float* __restrict__ al_s,
                            float* __restrict__ al_d) {
  __shared__ float ss[H * C];
  __shared__ float sd[H * C];
  int n = blockIdx.x, t = threadIdx.x;
  float v = h[(size_t)n * (H * C) + t];
  ss[t] = v * a_src[t];
  sd[t] = v * a_dst[t];
  __syncthreads();
  int c = t % C;
  for (int s = C / 2; s > 0; s >>= 1) {
    if (c < s) { ss[t] += ss[t + s]; sd[t] += sd[t + s]; }
    __syncthreads();
  }
  if (c == 0) {
    int head = t / C;
    al_s[n * H + head] = ss[t];
    al_d[n * H + head] = sd[t];
  }
}

__device__ __forceinline__ void edge_endpoints(const int* __restrict__ ei, int e,
                                               int& src, int& dst) {
  if (e < NE) { src = ei[e]; dst = ei[NE + e]; }
  else        { src = dst = e - NE; }          // appended self loops
}

template <int H>
__global__ void edge_max_kernel(const int* __restrict__ ei, const float* __restrict__ al_s,
                                const float* __restrict__ al_d, float* __restrict__ maxb) {
  int e = blockIdx.x * blockDim.x + threadIdx.x;
  if (e >= ET) return;
  int src, dst; edge_endpoints(ei, e, src, dst);
  #pragma unroll
  for (int hh = 0; hh < H; ++hh) {
    float ev = lrelu02(al_s[src * H + hh] + al_d[dst * H + hh]);
    atomicMaxF(&maxb[dst * H + hh], ev);
  }
}

template <int H>
__global__ void edge_sum_kernel(const int* __restrict__ ei, const float* __restrict__ al_s,
                                const float* __restrict__ al_d, const float* __restrict__ maxb,
                                float* __restrict__ sumb) {
  int e = blockIdx.x * blockDim.x + threadIdx.x;
  if (e >= ET) return;
  int src, dst; edge_endpoints(ei, e, src, dst);
  #pragma unroll
  for (int hh = 0; hh < H; ++hh) {
    float ev = lrelu02(al_s[src * H + hh] + al_d[dst * H + hh]);
    atomicAdd(&sumb[dst * H + hh], expf(ev - maxb[dst * H + hh]));
  }
}

// one wave per edge; lane handles (H*C)/32 contiguous channels (all in one head)
template <int H, int C>
__global__ __launch_bounds__(256) void edge_aggr_kernel(
    const int* __restrict__ ei, const float* __restrict__ hsrc,
    const float* __restrict__ al_s, const float* __restrict__ al_d,
    const float* __restrict__ maxb, const float* __restrict__ sumb,
    float* __restrict__ out) {
  const int HC = H * C, PER = HC / 32;
  int wave = threadIdx.x >> 5;
  int lane = threadIdx.x & 31;
  int e = blockIdx.x * 8 + wave;
  if (e >= ET) return;
  int src, dst; edge_endpoints(ei, e, src, dst);
  int head = (lane * PER) / C;
  float ev    = lrelu02(al_s[src * H + head] + al_d[dst * H + head]);
  float alpha = expf(ev - maxb[dst * H + head]) / (sumb[dst * H + head] + 1e-16f);
  const float* hs = hsrc + (size_t)src * HC + lane * PER;
  float*       op = out  + (size_t)dst * HC + lane * PER;
  #pragma unroll
  for (int j = 0; j < PER; ++j) atomicAdd(&op[j], alpha * hs[j]);
}

// ---------------- batch norm (stats + apply, ELU fused) ----------------
__global__ __launch_bounds__(128) void bn_stats_kernel(const float* __restrict__ h,
                                                       float* __restrict__ stats) {
  int c = threadIdx.x;
  int row0 = blockIdx.x * 512;
  int rend = min(row0 + 512, NN);
  float s = 0.f, ss = 0.f;
  for (int n = row0; n < rend; ++n) {
    float v = h[(size_t)n * 128 + c];
    s += v; ss += v * v;
  }
  atomicAdd(&stats[c], s);
  atomicAdd(&stats[128 + c], ss);
}

__global__ void bn_apply_kernel(float* __restrict__ h, const float* __restrict__ stats,
                                const float* __restrict__ g, const float* __restrict__ be) {
  size_t i = (size_t)blockIdx.x * blockDim.x + threadIdx.x;
  if (i >= (size_t)NN * 128) return;
  int c = (int)(i & 127);
  float mu  = stats[c] * (1.f / NN);
  float var = stats[128 + c] * (1.f / NN) - mu * mu;   // biased var, matches jnp.var
  float y = (h[i] - mu) * rsqrtf(var + EPS_BN) * g[c] + be[c];
  h[i] = eluf(y);
}

__global__ void bias_elu_kernel(float* __restrict__ h, const float* __restrict__ b) {
  size_t i = (size_t)blockIdx.x * blockDim.x + threadIdx.x;
  if (i >= (size_t)NN * 64) return;
  h[i] = eluf(h[i] + b[i & 63]);
}

// ---------------- global pooling (mean + max per graph) ----------------
__global__ void pool_kernel(const float* __restrict__ h, const int* __restrict__ batch,
                            float* __restrict__ psum, float* __restrict__ pmax,
                            float* __restrict__ counts) {
  size_t i = (size_t)blockIdx.x * blockDim.x + threadIdx.x;
  if (i >= (size_t)NN * 64) return;
  int n = (int)(i >> 6), c = (int)(i & 63);
  int g = batch[n];
  float v = h[i];
  atomicAdd(&psum[g * 64 + c], v);
  atomicMaxF(&pmax[g * 64 + c], v);
  if (c == 0) atomicAdd(&counts[g], 1.f);
}

// ---------------- classifier: Linear(128,64) -> ELU -> Linear(64,1) ----------------
__global__ __launch_bounds__(64) void mlp_kernel(const float* __restrict__ psum,
    const float* __restrict__ pmax, const float* __restrict__ counts,
    const float* __restrict__ w1, const float* __restrict__ b1,
    const float* __restrict__ w2, const float* __restrict__ b2,
    float* __restrict__ out) {
  __shared__ float hg[128];
  __shared__ float red[64];
  int g = blockIdx.x, j = threadIdx.x;
  float cnt = counts[g];
  float inv = 1.f / fmaxf(cnt, 1.f);
  hg[j]      = psum[g * 64 + j] * inv;
  hg[64 + j] = (cnt > 0.f) ? pmax[g * 64 + j] : 0.f;
  __syncthreads();
  float acc = b1[j];
  for (int k = 0; k < 128; ++k) acc += hg[k] * w1[k * 64 + j];
  red[j] = eluf(acc) * w2[j];
  __syncthreads();
  for (int s = 32; s > 0; s >>= 1) { if (j < s) red[j] += red[j + s]; __syncthreads(); }
  if (j == 0) out[g] = red[0] + b2[0];
}

extern "C" void kernel_launch(void* const* d_in, const int* in_sizes, int n_in,
                              void* d_out, int out_size, void* d_ws, size_t ws_size,
                              hipStream_t stream) {
  (void)in_sizes; (void)n_in; (void)out_size; (void)ws_size;
  const float* x   = (const float*)d_in[0];
  const int*   ei  = (const int*)d_in[1];
  const int*   bat = (const int*)d_in[2];
  const float* W1  = (const float*)d_in[3];
  const float* a1s = (const float*)d_in[4];
  const float* a1d = (const float*)d_in[5];
  // d_in[6] = b1 : cancels through batch-norm, never needed
  const float* g1  = (const float*)d_in[7];
  const float* be1 = (const float*)d_in[8];
  const float* W2  = (const float*)d_in[9];
  const float* a2s = (const float*)d_in[10];
  const float* a2d = (const float*)d_in[11];
  // d_in[12] = b2 : cancels through batch-norm
  const float* g2  = (const float*)d_in[13];
  const float* be2 = (const float*)d_in[14];
  const float* W3  = (const float*)d_in[15];
  const float* a3s = (const float*)d_in[16];
  const float* a3d = (const float*)d_in[17];
  const float* b3  = (const float*)d_in[18];
  const float* fw1 = (const float*)d_in[19];
  const float* fb1 = (const float*)d_in[20];
  const float* fw2 = (const float*)d_in[21];
  const float* fb2 = (const float*)d_in[22];
  float* out = (float*)d_out;

  float* ws    = (float*)d_ws;
  float* bufA  = ws;                              // NN*128 : linear-transformed feats
  float* bufB  = bufA  + (size_t)NN * 128;        // NN*128 : aggregation / layer output
  float* als   = bufB  + (size_t)NN * 128;        // NN*4
  float* ald   = als   + (size_t)NN * 4;          // NN*4
  float* maxb  = ald   + (size_t)NN * 4;          // NN*4
  float* sumb  = maxb  + (size_t)NN * 4;          // NN*4
  float* stats = sumb  + (size_t)NN * 4;          // 256
  float* psum  = stats + 256;                     // NG*64
  float* pmax  = psum  + (size_t)NG * 64;         // NG*64
  float* cnts  = pmax  + (size_t)NG * 64;         // NG

  auto fill = [&](float* p, float v, size_t n) {
    fill_kernel<<<(unsigned)((n + 255) / 256), 256, 0, stream>>>(p, v, (int)n);
  };
  const int EB  = (ET + 255) / 256;   // 1 thread / edge
  const int EBW = (ET + 7) / 8;       // 1 wave / edge (8 waves per 256-block)

  // ---------- conv1: 5 -> 4x32 concat ----------
  lin1_kernel<<<NN, 128, 0, stream>>>(x, W1, bufA);
  attn_kernel<4, 32><<<NN, 128, 0, stream>>>(bufA, a1s, a1d, als, ald);
  fill(maxb, -INFINITY, (size_t)NN * 4);
  fill(sumb, 0.f, (size_t)NN * 4);
  fill(bufB, 0.f, (size_t)NN * 128);
  edge_max_kernel<4><<<EB, 256, 0, stream>>>(ei, als, ald, maxb);
  edge_sum_kernel<4><<<EB, 256, 0, stream>>>(ei, als, ald, maxb, sumb);
  edge_aggr_kernel<4, 32><<<EBW, 256, 0, stream>>>(ei, bufA, als, ald, maxb, sumb, bufB);
  fill(stats, 0.f, 256);
  bn_stats_kernel<<<(NN + 511) / 512, 128, 0, stream>>>(bufB, stats);
  bn_apply_kernel<<<((size_t)NN * 128 + 255) / 256, 256, 0, stream>>>(bufB, stats, g1, be1);

  // ---------- conv2: 128 -> 4x32 concat (WMMA f32 GEMM) ----------
  wmma_gemm_kernel<<<dim3(NN / 16, 8), 32, 0, stream>>>(bufB, W2, bufA, 128, 128);
  attn_kernel<4, 32><<<NN, 128, 0, stream>>>(bufA, a2s, a2d, als, ald);
  fill(maxb, -INFINITY, (size_t)NN * 4);
  fill(sumb, 0.f, (size_t)NN * 4);
  fill(bufB, 0.f, (size_t)NN * 128);
  edge_max_kernel<4><<<EB, 256, 0, stream>>>(ei, als, ald, maxb);
  edge_sum_kernel<4><<<EB, 256, 0, stream>>>(ei, als, ald, maxb, sumb);
  edge_aggr_kernel<4, 32><<<EBW, 256, 0, stream>>>(ei, bufA, als, ald, maxb, sumb, bufB);
  fill(stats, 0.f, 256);
  bn_stats_kernel<<<(NN + 511) / 512, 128, 0, stream>>>(bufB, stats);
  bn_apply_kernel<<<((size_t)NN * 128 + 255) / 256, 256, 0, stream>>>(bufB, stats, g2, be2);

  // ---------- conv3: 128 -> 64, 1 head (WMMA f32 GEMM) ----------
  wmma_gemm_kernel<<<dim3(NN / 16, 4), 32, 0, stream>>>(bufB, W3, bufA, 128, 64);
  attn_kernel<1, 64><<<NN, 64, 0, stream>>>(bufA, a3s, a3d, als, ald);
  fill(maxb, -INFINITY, (size_t)NN);
  fill(sumb, 0.f, (size_t)NN);
  fill(bufB, 0.f, (size_t)NN * 64);
  edge_max_kernel<1><<<EB, 256, 0, stream>>>(ei, als, ald, maxb);
  edge_sum_kernel<1><<<EB, 256, 0, stream>>>(ei, als, ald, maxb, sumb);
  edge_aggr_kernel<1, 64><<<EBW, 256, 0, stream>>>(ei, bufA, als, ald, maxb, sumb, bufB);
  bias_elu_kernel<<<((size_t)NN * 64 + 255) / 256, 256, 0, stream>>>(bufB, b3);

  // ---------- pooling + classifier ----------
  fill(psum, 0.f, (size_t)NG * 64);
  fill(pmax, -INFINITY, (size_t)NG * 64);
  fill(cnts, 0.f, NG);
  pool_kernel<<<((size_t)NN * 64 + 255) / 256, 256, 0, stream>>>(bufB, bat, psum, pmax, cnts);
  mlp_kernel<<<NG, 64, 0, stream>>>(psum, pmax, cnts, fw1, fb1, fw2, fb2, out);
}